// GIN_32066225832278
// MI455X (gfx1250) — compile-verified
//
#include <hip/hip_runtime.h>

// GIN (3 layers, eval) on MI455X / gfx1250, wave32, fp32 WMMA (16x16x4).
// Rationale: scatter-add over edges is the bottleneck (~1.2 GB L2 traffic,
// ~300M atomics); GEMMs are only ~9.8 GFLOP, so we keep exact fp32 precision
// via V_WMMA_F32_16X16X4_F32 with register-resident weight fragments.

typedef __attribute__((ext_vector_type(2))) float v2f;
typedef __attribute__((ext_vector_type(8))) float v8f;

#define NN_NODES 50000
#define NN_EDGES 800000
#define HID 128
#define OUTC 16
#define NGRAPH 128
#define BN_EPS 1e-5f
#define MTILE 16
#define NROWTILES (NN_NODES / MTILE) // 3125 exactly

__global__ void zero_f32(float* __restrict__ p, int n) {
  int i = blockIdx.x * blockDim.x + threadIdx.x;
  int stride = gridDim.x * blockDim.x;
  for (; i < n; i += stride) p[i] = 0.0f;
}

// One wave per edge: lane loads float4 (32 lanes x 4 = 128 ch) of h[src],
// atomically accumulates into agg[dst].
__global__ void scatter_add(const float* __restrict__ h,
                            const int* __restrict__ src,
                            const int* __restrict__ dst,
                            float* __restrict__ agg) {
  int e = blockIdx.x * (blockDim.x >> 5) + (threadIdx.x >> 5);
  if (e >= NN_EDGES) return;
  int lane = threadIdx.x & 31;
  int s = src[e];
  int d = dst[e];
  const float4 v = ((const float4*)(h + (long)s * HID))[lane];
  float* a = agg + (long)d * HID + lane * 4;
  atomicAdd(a + 0, v.x);
  atomicAdd(a + 1, v.y);
  atomicAdd(a + 2, v.z);
  atomicAdd(a + 3, v.w);
}

// Persistent-block fused MLP: z = relu((h+agg)@W1+b1)@W2+b2; h' = relu(BN(z)).
// 8 waves per block; wave w owns output columns [16w, 16w+16).
// B-fragments for both GEMMs are held in registers for the whole kernel.
__global__ __launch_bounds__(256) void gin_mlp(
    const float* __restrict__ h, const float* __restrict__ agg,
    const float* __restrict__ W1, const float* __restrict__ b1,
    const float* __restrict__ W2, const float* __restrict__ b2,
    const float* __restrict__ gamma, const float* __restrict__ beta,
    const float* __restrict__ mean, const float* __restrict__ var,
    float* __restrict__ hout) {
  __shared__ float lds_in[MTILE * HID]; // 8 KB
  __shared__ float lds_z[MTILE * HID];  // 8 KB

  const int lane = threadIdx.x & 31;
  const int wv = threadIdx.x >> 5; // 0..7 -> N-tile
  const int nn = lane & 15;
  const int kh = lane >> 4; // 0/1: K-half within a K=4 step (A/B layout)
  const int col = wv * 16 + nn;

  // Per-lane epilogue constants (column is invariant across the 8 C VGPRs).
  const float b1v = b1[col];
  const float b2v = b2[col];
  const float invv = gamma[col] * rsqrtf(var[col] + BN_EPS);
  const float meanv = mean[col];
  const float betav = beta[col];

  // Register-resident B fragments: B(k,n) -> vgpr = k&1, lane = (k>>1 half)*16+n.
  v2f B1f[32], B2f[32];
#pragma unroll
  for (int ks = 0; ks < 32; ++ks) {
    int k = ks * 4 + kh * 2;
    B1f[ks].x = W1[(k + 0) * HID + col];
    B1f[ks].y = W1[(k + 1) * HID + col];
    B2f[ks].x = W2[(k + 0) * HID + col];
    B2f[ks].y = W2[(k + 1) * HID + col];
  }

  for (int t = blockIdx.x; t < NROWTILES; t += gridDim.x) {
    const long row0 = (long)t * MTILE;
    // Fused input staging: in = h + agg, 16x128 tile.
    for (int i = threadIdx.x; i < MTILE * HID; i += 256) {
      lds_in[i] = h[row0 * HID + i] + agg[row0 * HID + i];
    }
    __syncthreads();

    // GEMM1: C1 = in @ W1 ; 32 steps of K=4 fp32 WMMA.
    v8f c1 = {};
#pragma unroll
    for (int ks = 0; ks < 32; ++ks) {
      int k = ks * 4 + kh * 2;
      v2f a;
      a.x = lds_in[nn * HID + k + 0];
      a.y = lds_in[nn * HID + k + 1];
      c1 = __builtin_amdgcn_wmma_f32_16x16x4_f32(false, a, false, B1f[ks],
                                                 (short)0, c1, false, false);
    }
    // bias + ReLU, write z1 tile to LDS. C layout: row m = kh*8+r, col fixed.
#pragma unroll
    for (int r = 0; r < 8; ++r) {
      int m = kh * 8 + r;
      lds_z[m * HID + col] = fmaxf(c1[r] + b1v, 0.0f);
    }
    __syncthreads();

    // GEMM2: C2 = z1 @ W2.
    v8f c2 = {};
#pragma unroll
    for (int ks = 0; ks < 32; ++ks) {
      int k = ks * 4 + kh * 2;
      v2f a;
      a.x = lds_z[nn * HID + k + 0];
      a.y = lds_z[nn * HID + k + 1];
      c2 = __builtin_amdgcn_wmma_f32_16x16x4_f32(false, a, false, B2f[ks],
                                                 (short)0, c2, false, false);
    }
    // bias2 + BatchNorm(eval) + ReLU, store h'.
#pragma unroll
    for (int r = 0; r < 8; ++r) {
      int m = kh * 8 + r;
      float z = c2[r] + b2v;
      hout[(row0 + m) * HID + col] = fmaxf((z - meanv) * invv + betav, 0.0f);
    }
    __syncthreads();
  }
}

// Segment-sum pooling: one wave per node, atomic add into pooled[batch[n]].
__global__ void pool_add(const float* __restrict__ h,
                         const int* __restrict__ batch,
                         float* __restrict__ pooled) {
  int node = blockIdx.x * (blockDim.x >> 5) + (threadIdx.x >> 5);
  if (node >= NN_NODES) return;
  int lane = threadIdx.x & 31;
  int g = batch[node];
  const float4 v = ((const float4*)(h + (long)node * HID))[lane];
  float* p = pooled + (long)g * HID + lane * 4;
  atomicAdd(p + 0, v.x);
  atomicAdd(p + 1, v.y);
  atomicAdd(p + 2, v.z);
  atomicAdd(p + 3, v.w);
}

// Classifier: out[128x16] = pooled[128x128] @ Wc[128x16] + bc.
// One block, 8 waves; wave w computes rows [16w,16w+16), single N=16 tile.
__global__ __launch_bounds__(256) void classify(const float* __restrict__ pooled,
                                                const float* __restrict__ Wc,
                                                const float* __restrict__ bc,
                                                float* __restrict__ out) {
  const int lane = threadIdx.x & 31;
  const int wv = threadIdx.x >> 5; // M-tile 0..7
  const int nn = lane & 15;
  const int kh = lane >> 4;
  v8f c = {};
#pragma unroll
  for (int ks = 0; ks < 32; ++ks) {
    int k = ks * 4 + kh * 2;
    v2f a, b;
    a.x = pooled[(wv * 16 + nn) * HID + k + 0];
    a.y = pooled[(wv * 16 + nn) * HID + k + 1];
    b.x = Wc[(k + 0) * OUTC + nn];
    b.y = Wc[(k + 1) * OUTC + nn];
    c = __builtin_amdgcn_wmma_f32_16x16x4_f32(false, a, false, b, (short)0, c,
                                              false, false);
  }
  const float bcv = bc[nn];
#pragma unroll
  for (int r = 0; r < 8; ++r) {
    int m = wv * 16 + kh * 8 + r;
    out[m * OUTC + nn] = c[r] + bcv;
  }
}

extern "C" void kernel_launch(void* const* d_in, const int* in_sizes, int n_in,
                              void* d_out, int out_size, void* d_ws, size_t ws_size,
                              hipStream_t stream) {
  const float* x     = (const float*)d_in[0];
  const int*   edge  = (const int*)d_in[1];  // [2, E]: src row, then dst row
  const int*   batch = (const int*)d_in[2];
  const float* W1    = (const float*)d_in[3];
  const float* b1    = (const float*)d_in[4];
  const float* W2    = (const float*)d_in[5];
  const float* b2    = (const float*)d_in[6];
  const float* gam   = (const float*)d_in[7];
  const float* bet   = (const float*)d_in[8];
  const float* mean  = (const float*)d_in[9];
  const float* var   = (const float*)d_in[10];
  const float* Wc    = (const float*)d_in[11];
  const float* bc    = (const float*)d_in[12];
  float* out = (float*)d_out;

  const size_t FEAT = (size_t)NN_NODES * HID; // 6.4M floats
  float* agg    = (float*)d_ws;
  float* hA     = agg + FEAT;
  float* hB     = hA + FEAT;
  float* pooled = hB + FEAT;

  const int* src = edge;
  const int* dst = edge + NN_EDGES;

  const float* hcur = x;
  float* bufs[2] = {hA, hB};
  for (int layer = 0; layer < 3; ++layer) {
    zero_f32<<<2048, 256, 0, stream>>>(agg, (int)FEAT);
    scatter_add<<<(NN_EDGES + 7) / 8, 256, 0, stream>>>(hcur, src, dst, agg);
    float* hout = bufs[layer & 1];
    gin_mlp<<<512, 256, 0, stream>>>(
        hcur, agg,
        W1 + (size_t)layer * HID * HID, b1 + (size_t)layer * HID,
        W2 + (size_t)layer * HID * HID, b2 + (size_t)layer * HID,
        gam + (size_t)layer * HID, bet + (size_t)layer * HID,
        mean + (size_t)layer * HID, var + (size_t)layer * HID, hout);
    hcur = hout;
  }
  zero_f32<<<64, 256, 0, stream>>>(pooled, NGRAPH * HID);
  pool_add<<<(NN_NODES + 7) / 8, 256, 0, stream>>>(hcur, batch, pooled);
  classify<<<1, 256, 0, stream>>>(pooled, Wc, bc, out);
}